// GCNLayer_32435593019562
// MI455X (gfx1250) — compile-verified
//
#include <hip/hip_runtime.h>
#include <stdint.h>

// GCN layer: out = feature@W_self + in_deg^{-1/2} * (segsum_dst(feat[src]*e_w) @ W + b)
// where feat = feature * out_deg^{-1/2}.
// N=50000 nodes, E=800000 edges, F=256 in/out features, all fp32.

#define N_NODES 50000
#define N_EDGES 800000
#define F 256

typedef float v2f __attribute__((ext_vector_type(2)));
typedef float v8f __attribute__((ext_vector_type(8)));

// ---------------------------------------------------------------------------
// Async global->LDS copy (CDNA5): 16 bytes per lane, tracked by ASYNCcnt.
// lds_byte is the LDS byte address (low 32 bits of a generic pointer into a
// __shared__ array: flat LDS addresses truncate to addr[31:0] per the ISA
// aperture rules).
// ---------------------------------------------------------------------------
__device__ __forceinline__ void async_copy_b128(uint32_t lds_byte,
                                                const float* gaddr) {
  asm volatile("global_load_async_to_lds_b128 %0, %1, off"
               :: "v"(lds_byte), "v"(gaddr)
               : "memory");
}
__device__ __forceinline__ void wait_asynccnt0() {
  asm volatile("s_wait_asynccnt 0x0" ::: "memory");
}

// ---------------------------------------------------------------------------
// Stage 1: degree counting (out-degree over src, in-degree over dst)
// ---------------------------------------------------------------------------
__global__ __launch_bounds__(256) void deg_kernel(const int* __restrict__ src,
                                                  const int* __restrict__ dst,
                                                  float* __restrict__ deg_out,
                                                  float* __restrict__ deg_in,
                                                  int E) {
  int e = blockIdx.x * blockDim.x + threadIdx.x;
  if (e < E) {
    atomicAdd(deg_out + src[e], 1.0f);
    atomicAdd(deg_in + dst[e], 1.0f);
  }
}

// ---------------------------------------------------------------------------
// Stage 2: feat[i] = feature[i] * rsqrt(max(out_deg[i],1))   (float4 vectorized)
// ---------------------------------------------------------------------------
__global__ __launch_bounds__(256) void scale_kernel(const float* __restrict__ feature,
                                                    const float* __restrict__ deg_out,
                                                    float* __restrict__ feat,
                                                    int n) {
  int id = blockIdx.x * blockDim.x + threadIdx.x;   // n * 64 threads
  int i = id >> 6;
  int c = (id & 63) << 2;
  if (i < n) {
    float s = rsqrtf(fmaxf(deg_out[i], 1.0f));
    float4 v = *(const float4*)(feature + (size_t)i * F + c);
    float4 r;
    r.x = v.x * s; r.y = v.y * s; r.z = v.z * s; r.w = v.w * s;
    *(float4*)(feat + (size_t)i * F + c) = r;
  }
}

// ---------------------------------------------------------------------------
// Stage 3: edge scatter  agg[dst] += feat[src] * e_w
// One thread per (edge, 4-float chunk): float4 gather (L2-resident, 51MB table)
// + 4 global_atomic_add_f32 into the L2-resident accumulator.
// ---------------------------------------------------------------------------
__global__ __launch_bounds__(256) void scatter_kernel(const float* __restrict__ feat,
                                                      const float* __restrict__ e_w,
                                                      const int* __restrict__ src,
                                                      const int* __restrict__ dst,
                                                      float* __restrict__ agg,
                                                      int E) {
  long long id = (long long)blockIdx.x * blockDim.x + threadIdx.x;
  int e = (int)(id >> 6);
  int c = ((int)id & 63) << 2;
  if (e < E) {
    int s = src[e];
    int d = dst[e];
    float w = e_w[e];
    float4 v = *(const float4*)(feat + (size_t)s * F + c);
    float* o = agg + (size_t)d * F + c;
    atomicAdd(o + 0, v.x * w);
    atomicAdd(o + 1, v.y * w);
    atomicAdd(o + 2, v.z * w);
    atomicAdd(o + 3, v.w * w);
  }
}

// ---------------------------------------------------------------------------
// Stage 4: fused dual GEMM + epilogue via V_WMMA_F32_16X16X4_F32.
//   out[m,n] = sum_k feature[m,k]*Wself[k,n]
//            + rsqrt(max(in_deg[m],1)) * (sum_k agg[m,k]*W[k,n] + b[n])
// Block = 256 threads = 8 waves; blockIdx.x = N-tile (16 cols), blockIdx.y =
// group of 16 M-tiles; each wave register-blocks 2 M-tiles (32 rows) so every
// B fragment feeds 2 WMMAs. B panels (256x16 per matrix, 32 KB total) are
// staged via GLOBAL_LOAD_ASYNC_TO_LDS_B128 (ASYNCcnt) in a [k][16] layout
// that is LDS-bank conflict free for the WMMA read pattern (lane halves hit
// bank sets 32 apart). Partial last M-group is handled by clamping the tile
// index: duplicated waves compute identical values (deterministic), and EXEC
// stays all-ones at every WMMA as the ISA requires.
// ---------------------------------------------------------------------------
__global__ __launch_bounds__(256) void gemm_fused_kernel(
    const float* __restrict__ feature, const float* __restrict__ agg,
    const float* __restrict__ Wself, const float* __restrict__ Wn,
    const float* __restrict__ bias, const float* __restrict__ deg_in,
    float* __restrict__ out, int mtiles) {
  __shared__ float BsS[F * 16];
  __shared__ float BsN[F * 16];

  const int tid = threadIdx.x;
  const int n0 = blockIdx.x * 16;

  // Async-stage both 256x16 weight panels into LDS: 1024 b128 transfers per
  // matrix, 4 per thread. flat b128 index -> row k = flat>>2, col c = (flat&3)*4.
  const uint32_t ldsS = (uint32_t)(uintptr_t)&BsS[0];
  const uint32_t ldsN = (uint32_t)(uintptr_t)&BsN[0];
  for (int i = 0; i < 4; ++i) {
    int flat = i * 256 + tid;
    int k = flat >> 2;
    int c = (flat & 3) << 2;
    uint32_t off = (uint32_t)(k * 16 + c) * 4u;
    async_copy_b128(ldsS + off, Wself + (size_t)k * F + n0 + c);
    async_copy_b128(ldsN + off, Wn + (size_t)k * F + n0 + c);
  }
  wait_asynccnt0();
  __syncthreads();

  const int wave = tid >> 5;
  const int lane = tid & 31;
  int mt0 = blockIdx.y * 16 + wave;
  int mt1 = mt0 + 8;
  if (mt0 > mtiles - 1) mt0 = mtiles - 1;   // clamp (wave-uniform, branch-free)
  if (mt1 > mtiles - 1) mt1 = mtiles - 1;
  const int m0 = mt0 * 16;
  const int m1 = mt1 * 16;

  // WMMA f32 16x16x4 fragment indexing (ISA 7.12.2):
  //   A (16x4):  lane half selects K base {0,1} vs {2,3}; row = lane&15
  //   B (4x16):  same K split; col = lane&15
  //   C/D:       reg r -> row r + 8*(lane>>4); col = lane&15
  const int am = lane & 15;
  const int kb = (lane >> 4) * 2;
  const int half = lane >> 4;

  const float* As0 = feature + (size_t)(m0 + am) * F;
  const float* As1 = feature + (size_t)(m1 + am) * F;
  const float* An0 = agg + (size_t)(m0 + am) * F;
  const float* An1 = agg + (size_t)(m1 + am) * F;

  v8f accS0 = {0.f, 0.f, 0.f, 0.f, 0.f, 0.f, 0.f, 0.f};
  v8f accS1 = accS0;
  v8f accN0 = accS0;
  v8f accN1 = accS0;

  for (int k = 0; k < F; k += 4) {
    v2f aS0 = *(const v2f*)(As0 + k + kb);          // global_load_b64
    v2f aS1 = *(const v2f*)(As1 + k + kb);
    v2f aN0 = *(const v2f*)(An0 + k + kb);
    v2f aN1 = *(const v2f*)(An1 + k + kb);
    v2f bS, bN;
    bS.x = BsS[(k + kb) * 16 + am];                 // ds_load_2addr_b32
    bS.y = BsS[(k + kb + 1) * 16 + am];
    bN.x = BsN[(k + kb) * 16 + am];
    bN.y = BsN[(k + kb + 1) * 16 + am];
    accS0 = __builtin_amdgcn_wmma_f32_16x16x4_f32(false, aS0, false, bS,
                                                  (short)0, accS0, false, false);
    accS1 = __builtin_amdgcn_wmma_f32_16x16x4_f32(false, aS1, false, bS,
                                                  (short)0, accS1, false, false);
    accN0 = __builtin_amdgcn_wmma_f32_16x16x4_f32(false, aN0, false, bN,
                                                  (short)0, accN0, false, false);
    accN1 = __builtin_amdgcn_wmma_f32_16x16x4_f32(false, aN1, false, bN,
                                                  (short)0, accN1, false, false);
  }

  const float bn = bias[n0 + am];
  for (int r = 0; r < 8; ++r) {
    int ma = m0 + r + 8 * half;
    int mb = m1 + r + 8 * half;
    float inva = rsqrtf(fmaxf(deg_in[ma], 1.0f));
    float invb = rsqrtf(fmaxf(deg_in[mb], 1.0f));
    out[(size_t)ma * F + n0 + am] = accS0[r] + inva * (accN0[r] + bn);
    out[(size_t)mb * F + n0 + am] = accS1[r] + invb * (accN1[r] + bn);
  }
}

// ---------------------------------------------------------------------------
// Launch
// ---------------------------------------------------------------------------
extern "C" void kernel_launch(void* const* d_in, const int* in_sizes, int n_in,
                              void* d_out, int out_size, void* d_ws, size_t ws_size,
                              hipStream_t stream) {
  // setup_inputs order: feature, e_w, snorm_n, snorm_e, W_self, W, b, src, dst
  const float* feature = (const float*)d_in[0];
  const float* e_w     = (const float*)d_in[1];
  const float* Wself   = (const float*)d_in[4];
  const float* Wn      = (const float*)d_in[5];
  const float* bias    = (const float*)d_in[6];
  const int*   src     = (const int*)d_in[7];
  const int*   dst     = (const int*)d_in[8];
  float* out = (float*)d_out;

  // Workspace layout: [deg_out N][deg_in N][feat N*F][agg N*F] (~103 MB)
  float* ws      = (float*)d_ws;
  float* deg_out = ws;
  float* deg_in  = ws + N_NODES;
  float* feat    = ws + 2 * N_NODES;
  float* agg     = feat + (size_t)N_NODES * F;

  // Zero accumulators every call (workspace is poisoned once, never restored).
  hipMemsetAsync(deg_out, 0, (size_t)2 * N_NODES * sizeof(float), stream);
  hipMemsetAsync(agg, 0, (size_t)N_NODES * F * sizeof(float), stream);

  deg_kernel<<<(N_EDGES + 255) / 256, 256, 0, stream>>>(src, dst, deg_out, deg_in, N_EDGES);

  scale_kernel<<<(N_NODES * 64 + 255) / 256, 256, 0, stream>>>(feature, deg_out, feat, N_NODES);

  long long sthreads = (long long)N_EDGES * 64;
  scatter_kernel<<<(int)((sthreads + 255) / 256), 256, 0, stream>>>(feat, e_w, src, dst, agg, N_EDGES);

  const int mtiles = N_NODES / 16;  // 3125 exactly (50000 % 16 == 0)
  dim3 grid(F / 16, (mtiles + 15) / 16);   // 16 x 196
  gemm_fused_kernel<<<grid, 256, 0, stream>>>(feature, agg, Wself, Wn, bias, deg_in, out, mtiles);

  // Second tuple output: e_w passthrough.
  hipMemcpyAsync(out + (size_t)N_NODES * F, e_w, (size_t)N_EDGES * sizeof(float),
                 hipMemcpyDeviceToDevice, stream);
}